// GAT_47064251629737
// MI455X (gfx1250) — compile-verified
//
#include <hip/hip_runtime.h>

#define D_IN  128
#define HEADS 8
#define HID   32
#define HS1   (HEADS * HID)   // 256
#define NEG_SLOPE 0.2f

typedef __attribute__((ext_vector_type(2))) float v2f;
typedef __attribute__((ext_vector_type(8))) float v8f;

// ---------------------------------------------------------------------------
// WMMA f32 GEMM: C[M,NN] = A[M,K] @ B[K,NN].
// K, NN are compile-time so all strided loads fold into immediate offsets.
// One wave computes a 16x32 output strip (two 16x16 tiles sharing the A frag).
// ---------------------------------------------------------------------------
template <int K, int NN>
__global__ __launch_bounds__(256) void gemm_wmma_f32_t(
    const float* __restrict__ A, const float* __restrict__ B,
    float* __restrict__ C, int M, int totalWaves)
{
  constexpr int NPAIR = NN / 32;   // 16x32 strips per row of tiles
  int wave = blockIdx.x * (blockDim.x >> 5) + (threadIdx.x >> 5);
  if (wave >= totalWaves) return;
  int mt = wave / NPAIR, np = wave - mt * NPAIR;
  int m0 = mt << 4, n0 = np << 5;

  int lane  = threadIdx.x & 31;
  int row   = lane & 15;          // A row / B,C column within tile
  int koff  = (lane >> 4) << 1;   // lanes 16-31 hold K+2,K+3
  int half8 = (lane >> 4) << 3;   // lanes 16-31 hold C rows 8..15

  int gm   = m0 + row;
  int arow = gm < M ? gm : (M - 1);            // clamp; store is guarded
  const float* Ap  = A + (size_t)arow * K + koff;
  const float* Bp0 = B + (size_t)koff * NN + n0 + row;   // tile 0 column
  const float* Bp1 = Bp0 + 16;                           // tile 1 column

  v8f acc0 = {}, acc1 = {};
#pragma unroll 8
  for (int k = 0; k < K; k += 4) {
    v2f a, b0, b1;
    a.x  = Ap[k];
    a.y  = Ap[k + 1];
    b0.x = Bp0[k * NN];
    b0.y = Bp0[k * NN + NN];
    b1.x = Bp1[k * NN];
    b1.y = Bp1[k * NN + NN];
    acc0 = __builtin_amdgcn_wmma_f32_16x16x4_f32(false, a, false, b0,
                                                 (short)0, acc0, false, false);
    acc1 = __builtin_amdgcn_wmma_f32_16x16x4_f32(false, a, false, b1,
                                                 (short)0, acc1, false, false);
  }
#pragma unroll
  for (int r = 0; r < 8; ++r) {
    int orow = m0 + half8 + r;
    if (orow < M) {
      C[(size_t)orow * NN + n0 + row]      = acc0[r];
      C[(size_t)orow * NN + n0 + 16 + row] = acc1[r];
    }
  }
}

// ---------------------------------------------------------------------------
// Fold attention vectors into the weight matrices:
//   weff[d,h] = sum_c W[d, h*HID+c] * a[h,c]     (exact identity)
// ---------------------------------------------------------------------------
__global__ void weff1_kernel(const float* __restrict__ Ws, const float* __restrict__ Wd,
                             const float* __restrict__ as, const float* __restrict__ ad,
                             float* __restrict__ es, float* __restrict__ ed)
{
  int idx = blockIdx.x * blockDim.x + threadIdx.x;
  if (idx >= D_IN * HEADS) return;
  int d = idx >> 3, h = idx & 7;
  float ss = 0.f, sd = 0.f;
#pragma unroll 8
  for (int c = 0; c < HID; ++c) {
    ss += Ws[d * HS1 + h * HID + c] * as[h * HID + c];
    sd += Wd[d * HS1 + h * HID + c] * ad[h * HID + c];
  }
  es[d * HEADS + h] = ss;
  ed[d * HEADS + h] = sd;
}

__global__ void weff2_kernel(const float* __restrict__ Ws, const float* __restrict__ Wd,
                             const float* __restrict__ as, const float* __restrict__ ad,
                             float* __restrict__ es, float* __restrict__ ed)
{
  int d = blockIdx.x * blockDim.x + threadIdx.x;
  if (d >= HS1) return;
  float ss = 0.f, sd = 0.f;
#pragma unroll 8
  for (int c = 0; c < HID; ++c) {
    ss += Ws[d * HID + c] * as[c];
    sd += Wd[d * HID + c] * ad[c];
  }
  es[d] = ss;
  ed[d] = sd;
}

// alpha_s / alpha_d per node (layer 1): [N,8] = x[N,128] @ weff[128,8]
__global__ void alpha1_kernel(const float* __restrict__ x,
                              const float* __restrict__ es, const float* __restrict__ ed,
                              float* __restrict__ ao, float* __restrict__ bo, int N)
{
  int idx = blockIdx.x * blockDim.x + threadIdx.x;
  if (idx >= N * HEADS) return;
  int n = idx >> 3, h = idx & 7;
  const float* xr = x + (size_t)n * D_IN;
  float ss = 0.f, sd = 0.f;
#pragma unroll 4
  for (int d = 0; d < D_IN; ++d) {
    float xv = xr[d];
    ss += xv * es[d * HEADS + h];
    sd += xv * ed[d * HEADS + h];
  }
  ao[idx] = ss;
  bo[idx] = sd;
}

// alpha per node (layer 2, heads==1): h[N,256] @ weff[256]
__global__ void alpha2_kernel(const float* __restrict__ hbuf,
                              const float* __restrict__ es, const float* __restrict__ ed,
                              float* __restrict__ ao, float* __restrict__ bo, int N)
{
  int n = blockIdx.x * blockDim.x + threadIdx.x;
  if (n >= N) return;
  const float* hr = hbuf + (size_t)n * HS1;
  float ss = 0.f, sd = 0.f;
#pragma unroll 4
  for (int d = 0; d < HS1; ++d) {
    float hv = hr[d];
    ss += hv * es[d];
    sd += hv * ed[d];
  }
  ao[n] = ss;
  bo[n] = sd;
}

// Seed accumulator with broadcast bias (so "+b" is free); mod is a power of 2.
__global__ void bias_init_kernel(float* __restrict__ out, const float* __restrict__ b,
                                 int total, int mod)
{
  int idx = blockIdx.x * blockDim.x + threadIdx.x;
  if (idx < total) out[idx] = b[idx & (mod - 1)];
}

__global__ void relu_kernel(float* __restrict__ buf, int total)
{
  int idx = blockIdx.x * blockDim.x + threadIdx.x;
  if (idx < total) buf[idx] = fmaxf(buf[idx], 0.f);
}

// ---------------------------------------------------------------------------
// Segment softmax over edges: monotone uint encoding for exact float atomicMax
// ---------------------------------------------------------------------------
__device__ __forceinline__ unsigned enc_f32(float v) {
  unsigned u = __float_as_uint(v);
  return (int)u >= 0 ? (u | 0x80000000u) : ~u;
}
__device__ __forceinline__ float dec_f32(unsigned u) {
  return (u & 0x80000000u) ? __uint_as_float(u ^ 0x80000000u) : __uint_as_float(~u);
}

__global__ void edge_max_kernel(const int* __restrict__ src, const int* __restrict__ dst,
                                const float* __restrict__ as, const float* __restrict__ ad,
                                unsigned* __restrict__ maxbuf, int E, int heads)
{
  int idx = blockIdx.x * blockDim.x + threadIdx.x;
  if (idx >= E * heads) return;
  int e = idx / heads, h = idx - e * heads;
  int d = dst[e];
  float v = as[src[e] * heads + h] + ad[d * heads + h];
  v = v >= 0.f ? v : NEG_SLOPE * v;
  atomicMax(&maxbuf[d * heads + h], enc_f32(v));
}

__global__ void edge_exp_kernel(const int* __restrict__ src, const int* __restrict__ dst,
                                const float* __restrict__ as, const float* __restrict__ ad,
                                const unsigned* __restrict__ maxbuf,
                                float* __restrict__ ee, float* __restrict__ denom,
                                int E, int heads)
{
  int idx = blockIdx.x * blockDim.x + threadIdx.x;
  if (idx >= E * heads) return;
  int e = idx / heads, h = idx - e * heads;
  int d = dst[e];
  float v = as[src[e] * heads + h] + ad[d * heads + h];
  v = v >= 0.f ? v : NEG_SLOPE * v;
  unsigned m = maxbuf[d * heads + h];
  float mf = (m == 0u) ? 0.f : dec_f32(m);   // empty segment -> 0 (matches ref)
  float ex = __expf(v - mf);
  ee[idx] = ex;
  atomicAdd(&denom[d * heads + h], ex);
}

// Layer-1 aggregate: one 256-thread block per edge; t -> (head=t/32, ch=t%32)
__global__ __launch_bounds__(256) void edge_agg1_kernel(
    const int* __restrict__ src, const int* __restrict__ dst,
    const float* __restrict__ hs, const float* __restrict__ ee,
    const float* __restrict__ denom, float* __restrict__ out, int E)
{
  int e = blockIdx.x;
  if (e >= E) return;
  int t = threadIdx.x;
  int h = t >> 5;                         // wave-uniform
  int s = src[e], d = dst[e];
  float alpha = ee[(size_t)e * HEADS + h] / (denom[d * HEADS + h] + 1e-16f);
  atomicAdd(&out[(size_t)d * HS1 + t], hs[(size_t)s * HS1 + t] * alpha);
}

// Layer-2 aggregate: 32 channels per edge, 8 edges per block
__global__ void edge_agg2_kernel(const int* __restrict__ src, const int* __restrict__ dst,
                                 const float* __restrict__ hs, const float* __restrict__ ee,
                                 const float* __restrict__ denom, float* __restrict__ out,
                                 int E)
{
  int idx = blockIdx.x * blockDim.x + threadIdx.x;
  if (idx >= E * HID) return;
  int e = idx >> 5, c = idx & 31;
  int s = src[e], d = dst[e];
  float alpha = ee[e] / (denom[d] + 1e-16f);
  atomicAdd(&out[(size_t)d * HID + c], hs[(size_t)s * HID + c] * alpha);
}

// ---------------------------------------------------------------------------
extern "C" void kernel_launch(void* const* d_in, const int* in_sizes, int n_in,
                              void* d_out, int out_size, void* d_ws, size_t ws_size,
                              hipStream_t stream)
{
  const float* x   = (const float*)d_in[0];
  const int*   ei  = (const int*)d_in[1];      // int32 (JAX x64 disabled)
  const float* W1s = (const float*)d_in[2];
  const float* W1d = (const float*)d_in[3];
  const float* a1s = (const float*)d_in[4];
  const float* a1d = (const float*)d_in[5];
  const float* b1  = (const float*)d_in[6];
  const float* W2s = (const float*)d_in[7];
  const float* W2d = (const float*)d_in[8];
  const float* a2s = (const float*)d_in[9];
  const float* a2d = (const float*)d_in[10];
  const float* b2  = (const float*)d_in[11];

  const int N = in_sizes[0] / D_IN;
  const int E = in_sizes[1] / 2;
  const int* src = ei;
  const int* dst = ei + E;

  // ---- workspace carve-up (256B aligned) ----
  char* w = (char*)d_ws;
  auto alloc = [&](size_t bytes) -> char* {
    char* p = w;
    w += (bytes + 255) & ~(size_t)255;
    return p;
  };
  float*    hs1    = (float*)   alloc((size_t)N * HS1 * 4);
  float*    out1   = (float*)   alloc((size_t)N * HS1 * 4);  // becomes h after relu
  float*    as1    = (float*)   alloc((size_t)N * HEADS * 4);
  float*    ad1    = (float*)   alloc((size_t)N * HEADS * 4);
  unsigned* max1   = (unsigned*)alloc((size_t)N * HEADS * 4);
  float*    den1   = (float*)   alloc((size_t)N * HEADS * 4);
  float*    ee1    = (float*)   alloc((size_t)E * HEADS * 4);
  float*    weff1s = (float*)   alloc((size_t)D_IN * HEADS * 4);
  float*    weff1d = (float*)   alloc((size_t)D_IN * HEADS * 4);
  float*    hs2    = (float*)   alloc((size_t)N * HID * 4);
  float*    as2    = (float*)   alloc((size_t)N * 4);
  float*    ad2    = (float*)   alloc((size_t)N * 4);
  unsigned* max2   = (unsigned*)alloc((size_t)N * 4);
  float*    den2   = (float*)   alloc((size_t)N * 4);
  float*    ee2    = (float*)   alloc((size_t)E * 4);
  float*    weff2s = (float*)   alloc((size_t)HS1 * 4);
  float*    weff2d = (float*)   alloc((size_t)HS1 * 4);

  hipMemsetAsync(max1, 0, (size_t)N * HEADS * 4, stream);
  hipMemsetAsync(den1, 0, (size_t)N * HEADS * 4, stream);
  hipMemsetAsync(max2, 0, (size_t)N * 4, stream);
  hipMemsetAsync(den2, 0, (size_t)N * 4, stream);

  // ---- layer 1 ----
  weff1_kernel<<<(D_IN * HEADS + 255) / 256, 256, 0, stream>>>(W1s, W1d, a1s, a1d,
                                                               weff1s, weff1d);
  weff2_kernel<<<(HS1 + 255) / 256, 256, 0, stream>>>(W2s, W2d, a2s, a2d,
                                                      weff2s, weff2d);

  int mtiles = (N + 15) / 16;
  int waves1 = mtiles * (HS1 / 32);
  gemm_wmma_f32_t<D_IN, HS1><<<(waves1 + 7) / 8, 256, 0, stream>>>(x, W1s, hs1,
                                                                   N, waves1);

  alpha1_kernel<<<(N * HEADS + 255) / 256, 256, 0, stream>>>(x, weff1s, weff1d,
                                                             as1, ad1, N);
  bias_init_kernel<<<(N * HS1 + 255) / 256, 256, 0, stream>>>(out1, b1, N * HS1, HS1);

  edge_max_kernel<<<(E * HEADS + 255) / 256, 256, 0, stream>>>(src, dst, as1, ad1,
                                                               max1, E, HEADS);
  edge_exp_kernel<<<(E * HEADS + 255) / 256, 256, 0, stream>>>(src, dst, as1, ad1,
                                                               max1, ee1, den1, E, HEADS);
  edge_agg1_kernel<<<E, 256, 0, stream>>>(src, dst, hs1, ee1, den1, out1, E);

  relu_kernel<<<(N * HS1 + 255) / 256, 256, 0, stream>>>(out1, N * HS1);

  // ---- layer 2 ----
  int waves2 = mtiles * (HID / 32);
  gemm_wmma_f32_t<HS1, HID><<<(waves2 + 7) / 8, 256, 0, stream>>>(out1, W2s, hs2,
                                                                  N, waves2);

  alpha2_kernel<<<(N + 255) / 256, 256, 0, stream>>>(out1, weff2s, weff2d, as2, ad2, N);

  float* outF = (float*)d_out;
  bias_init_kernel<<<(N * HID + 255) / 256, 256, 0, stream>>>(outF, b2, N * HID, HID);

  edge_max_kernel<<<(E + 255) / 256, 256, 0, stream>>>(src, dst, as2, ad2, max2, E, 1);
  edge_exp_kernel<<<(E + 255) / 256, 256, 0, stream>>>(src, dst, as2, ad2, max2,
                                                       ee2, den2, E, 1);
  edge_agg2_kernel<<<(E * HID + 255) / 256, 256, 0, stream>>>(src, dst, hs2, ee2,
                                                              den2, outF, E);
}